// LightGCN_7395933684090
// MI455X (gfx1250) — compile-verified
//
#include <hip/hip_runtime.h>
#include <hip/hip_bf16.h>
#include <stdint.h>

typedef float v8f __attribute__((ext_vector_type(8)));
typedef float v2f __attribute__((ext_vector_type(2)));

#define GDIM 64    // embedding dim, fixed by the reference
#define CHUNK 256  // edges staged per block per iteration

// ---------------------------------------------------------------------------
// CDNA5 helpers (inline asm over paths the builtins don't cover on ROCm 7.2)
// ---------------------------------------------------------------------------
__device__ __forceinline__ uint32_t lds_addr_of(const void* p) {
  // generic(shared) pointer: low 32 bits == LDS byte offset (aperture low word is 0)
  return (uint32_t)(uintptr_t)p;
}

__device__ __forceinline__ void async_load_b32(uint32_t lds, uint64_t gbase, uint32_t goff) {
  // GVS form: copies 4 bytes from the global address gbase plus goff to the LDS address in lds; tracked by ASYNCcnt
  asm volatile("global_load_async_to_lds_b32 %0, %1, %2"
               :: "v"(lds), "v"(goff), "s"(gbase) : "memory");
}

__device__ __forceinline__ void atomic_add_f32_noret(uint64_t gbase, uint32_t voff, float v,
                                                     int imm4 /*0 or 1: +4B*/) {
  if (imm4)
    asm volatile("global_atomic_add_f32 %0, %1, %2 offset:4"
                 :: "v"(voff), "v"(v), "s"(gbase) : "memory");
  else
    asm volatile("global_atomic_add_f32 %0, %1, %2"
                 :: "v"(voff), "v"(v), "s"(gbase) : "memory");
}

// ---------------------------------------------------------------------------
// Zero-fill (float4 vectorized)
// ---------------------------------------------------------------------------
__global__ __launch_bounds__(256) void zero_kernel(float4* __restrict__ p, long long n4) {
  long long i = (long long)blockIdx.x * blockDim.x + threadIdx.x;
  const long long stride = (long long)gridDim.x * blockDim.x;
  float4 z; z.x = z.y = z.z = z.w = 0.0f;
  for (; i < n4; i += stride) p[i] = z;
}

// ---------------------------------------------------------------------------
// Scatter SpMM: out[row[e]] += ew[e] * h[col[e]]
// Edge metadata (row/col/w) is double-buffer staged into LDS with
// global_load_async_to_lds_b32 (ASYNCcnt), a full chunk ahead of use.
// Each wave owns one edge at a time: lane l handles dims {2l, 2l+1}:
// one global_load_b64 gather + two no-return global_atomic_add_f32.
// ---------------------------------------------------------------------------
__global__ __launch_bounds__(256) void spmm_scatter(
    const float* __restrict__ h, const float* __restrict__ ew,
    const int* __restrict__ row, const int* __restrict__ col,
    float* __restrict__ out, int E) {
  __shared__ int   sRow[2][CHUNK];
  __shared__ int   sCol[2][CHUNK];
  __shared__ float sW  [2][CHUNK];

  const int tid  = threadIdx.x;
  const int lane = tid & 31;
  const int wave = __builtin_amdgcn_readfirstlane(tid >> 5);

  const uint64_t rowB = (uint64_t)(uintptr_t)row;
  const uint64_t colB = (uint64_t)(uintptr_t)col;
  const uint64_t ewB  = (uint64_t)(uintptr_t)ew;
  const uint64_t outB = (uint64_t)(uintptr_t)out;

  const long long stride = (long long)gridDim.x * CHUNK;
  long long base = (long long)blockIdx.x * CHUNK;

  // stage chunk 0 into buffer 0
  if (base < E) {
    long long idx = base + tid;
    uint32_t goff = (uint32_t)((idx < E ? idx : (long long)E - 1) * 4);  // clamp, keep EXEC full
    async_load_b32(lds_addr_of(&sRow[0][tid]), rowB, goff);
    async_load_b32(lds_addr_of(&sCol[0][tid]), colB, goff);
    async_load_b32(lds_addr_of(&sW  [0][tid]), ewB,  goff);
  }

  int buf = 0;
  for (; base < E; base += stride, buf ^= 1) {
    const long long nxt = base + stride;
    const bool more = nxt < E;
    if (more) {
      long long idx = nxt + tid;
      uint32_t goff = (uint32_t)((idx < E ? idx : (long long)E - 1) * 4);
      async_load_b32(lds_addr_of(&sRow[buf ^ 1][tid]), rowB, goff);
      async_load_b32(lds_addr_of(&sCol[buf ^ 1][tid]), colB, goff);
      async_load_b32(lds_addr_of(&sW  [buf ^ 1][tid]), ewB,  goff);
      asm volatile("s_wait_asynccnt 3" ::: "memory");   // oldest 3 (this buffer) done
    } else {
      asm volatile("s_wait_asynccnt 0" ::: "memory");
    }
    __syncthreads();  // all waves' staging visible

    const int nE = (E - base < CHUNK) ? (int)(E - base) : CHUNK;
    for (int j = wave; j < nE; j += 8) {
      const int   r = sRow[buf][j];
      const int   c = sCol[buf][j];
      const float w = sW  [buf][j];
      const float2 v = ((const float2*)(h + (size_t)c * GDIM))[lane];
      const uint32_t voff = (uint32_t)r * (GDIM * 4u) + (uint32_t)lane * 8u;
      atomic_add_f32_noret(outB, voff, v.x * w, 0);
      atomic_add_f32_noret(outB, voff, v.y * w, 1);
    }
    __syncthreads();  // WAR: done reading before this buffer is re-staged
  }
}

// ---------------------------------------------------------------------------
// Combine: out = (emb + h1 + h2) / 3, one 16x16 tile per wave, computed as
// D = (1/3 * I16) x T via four V_WMMA_F32_16X16X4_F32 k-steps.
// (layouts per CDNA5 ISA 7.12.2; verified good codegen in round 1)
// ---------------------------------------------------------------------------
__global__ __launch_bounds__(256) void combine_wmma(
    const float* __restrict__ emb, const float* __restrict__ h1,
    float* __restrict__ h2out, int nTiles) {
  const int lane = threadIdx.x & 31;
  const int t = blockIdx.x * 8 + (threadIdx.x >> 5);   // tile id (wave-uniform)
  if (t >= nTiles) return;                             // uniform branch, EXEC stays full

  const int tc = t & 3;            // 64/16 = 4 tile columns
  const int tr = t >> 2;
  const int colBase = tc * 16 + (lane & 15);
  const long long rowBase = (long long)tr * 16;
  const int hi = (lane >= 16) ? 1 : 0;
  const int m  = lane & 15;
  const float third = 1.0f / 3.0f;

  v8f acc = {0.f, 0.f, 0.f, 0.f, 0.f, 0.f, 0.f, 0.f};

  #pragma unroll
  for (int k = 0; k < 4; ++k) {
    const int r0 = 4 * k + (hi ? 2 : 0);
    const int r1 = r0 + 1;
    const long long i0 = (rowBase + r0) * GDIM + colBase;
    const long long i1 = (rowBase + r1) * GDIM + colBase;
    v2f b;
    b.x = emb[i0] + h1[i0] + h2out[i0];
    b.y = emb[i1] + h1[i1] + h2out[i1];
    v2f a;
    a.x = (m == (4 * k + (hi ? 2 : 0))) ? third : 0.0f;
    a.y = (m == (4 * k + (hi ? 3 : 1))) ? third : 0.0f;
    acc = __builtin_amdgcn_wmma_f32_16x16x4_f32(
        false, a, false, b, (short)0, acc, false, false);
  }

  const int rOff = hi ? 8 : 0;
  #pragma unroll
  for (int v = 0; v < 8; ++v) {
    const long long idx = (rowBase + v + rOff) * GDIM + colBase;
    h2out[idx] = acc[v];
  }
}

// Scalar tail for row counts not divisible by 16 (not hit for N=100000).
__global__ __launch_bounds__(256) void tail_scale(
    const float* __restrict__ emb, const float* __restrict__ h1,
    float* __restrict__ h2out, long long start, long long end) {
  long long i = start + (long long)blockIdx.x * blockDim.x + threadIdx.x;
  const long long stride = (long long)gridDim.x * blockDim.x;
  for (; i < end; i += stride)
    h2out[i] = (emb[i] + h1[i] + h2out[i]) * (1.0f / 3.0f);
}

// ---------------------------------------------------------------------------
extern "C" void kernel_launch(void* const* d_in, const int* in_sizes, int n_in,
                              void* d_out, int out_size, void* d_ws, size_t ws_size,
                              hipStream_t stream) {
  const float* emb = (const float*)d_in[0];
  const float* ew  = (const float*)d_in[1];
  const int*   ei  = (const int*)d_in[2];   // [2, E] int32
  const int E      = in_sizes[1];
  const long long ND = in_sizes[0];         // N * 64
  const int N      = (int)(ND / GDIM);
  const int* row = ei;
  const int* col = ei + E;

  float* h1  = (float*)d_ws;                // layer-1 output (25.6 MB)
  float* out = (float*)d_out;               // layer-2 output, then final result

  const int zb = 2048, sb = 2048;

  // h1 = A @ emb
  zero_kernel <<<zb, 256, 0, stream>>>((float4*)h1, ND / 4);
  spmm_scatter<<<sb, 256, 0, stream>>>(emb, ew, row, col, h1, E);
  // out = A @ h1
  zero_kernel <<<zb, 256, 0, stream>>>((float4*)out, ND / 4);
  spmm_scatter<<<sb, 256, 0, stream>>>(h1, ew, row, col, out, E);
  // out = (emb + h1 + out) / 3   (WMMA tile scale)
  const int fullRowTiles = N / 16;
  const int nTiles = fullRowTiles * (GDIM / 16);
  if (nTiles > 0)
    combine_wmma<<<(nTiles + 7) / 8, 256, 0, stream>>>(emb, h1, out, nTiles);
  const long long tailStart = (long long)fullRowTiles * 16 * GDIM;
  if (tailStart < ND)
    tail_scale<<<256, 256, 0, stream>>>(emb, h1, out, tailStart, ND);
}